// FixedRadiusSearch_21603685499087
// MI455X (gfx1250) — compile-verified
//
#include <hip/hip_runtime.h>

typedef __attribute__((ext_vector_type(2))) float v2f;
typedef __attribute__((ext_vector_type(8))) float v8f;

#define NPTS 8192
#define NQRY 8192
#define KCAP 128

// One wave (32 lanes) owns a tile of 16 queries and sweeps all 8192 points in
// 16-point chunks. Each chunk: one V_WMMA_F32_16X16X4_F32 produces all 256
// (-2 * q.p) values; epilogue forms d2 = |q|^2 + |p|^2 + c, masks against r^2,
// and stream-compacts per query row via wave32 ballot + popcount.
// Point loads are software-pipelined one chunk ahead so the epilogue hides
// the VMEM latency instead of stalling on s_wait_loadcnt each iteration.
__global__ __launch_bounds__(256) void frs_search_kernel(
    const float* __restrict__ points,
    const float* __restrict__ queries,
    const float* __restrict__ radius_p,
    float* __restrict__ out_idx,
    float* __restrict__ out_dist,
    int*   __restrict__ counts)
{
    const int lane      = threadIdx.x & 31;
    const int waveInBlk = threadIdx.x >> 5;
    const int tile      = blockIdx.x * 8 + waveInBlk;   // 16 queries per tile
    const int col       = lane & 15;
    const bool hi       = lane >= 16;
    const int qrow_base = tile * 16;

    const float r  = radius_p[0];
    const float r2 = r * r;

    // Load this lane's query (M = col); both halves load same query -> L0 hit.
    const int q = qrow_base + col;
    const float qx = queries[3 * q + 0];
    const float qy = queries[3 * q + 1];
    const float qz = queries[3 * q + 2];
    const float q2full = qx * qx + qy * qy + qz * qz;

    // A operand (16x4 f32): lanes 0-15 hold K={0,1}; lanes 16-31 hold K={2,3}.
    v2f a;
    a.x = hi ? qz : qx;
    a.y = hi ? 0.0f : qy;

    // Per-C-row |q|^2: VGPR v covers row v (low half) and row v+8 (high half).
    float qq[8];
#pragma unroll
    for (int v = 0; v < 8; ++v)
        qq[v] = __shfl(q2full, hi ? (v + 8) : v, 32);

    // Loop-invariant epilogue helpers.
    const unsigned shamt      = hi ? 16u : 0u;          // ballot half select
    const unsigned lanebelow  = (1u << col) - 1u;       // bits strictly below col
    unsigned rowoff[8];                                  // row * KCAP (unsigned
#pragma unroll                                           //  -> scale_offset path)
    for (int v = 0; v < 8; ++v)
        rowoff[v] = (unsigned)(qrow_base + v + (hi ? 8 : 0)) * KCAP;

    int cnt[8];
#pragma unroll
    for (int v = 0; v < 8; ++v) cnt[v] = 0;

    // Prefetch chunk 0.
    float px = points[3 * col + 0];
    float py = points[3 * col + 1];
    float pz = points[3 * col + 2];

    for (int j0 = 0; j0 < NPTS; j0 += 16) {
        const int j = j0 + col;

        const float cpx = px, cpy = py, cpz = pz;

        // Issue next chunk's load now; consumed only after the epilogue.
        const int j0n = (j0 + 16 < NPTS) ? (j0 + 16) : 0;  // tail: dummy reload
        const int jn  = j0n + col;
        px = points[3 * jn + 0];
        py = points[3 * jn + 1];
        pz = points[3 * jn + 2];

        const float p2 = cpx * cpx + cpy * cpy + cpz * cpz;

        // B operand (4x16 f32), -2 folded in: accumulator = -2 * q.p
        v2f b;
        b.x = hi ? (-2.0f * cpz) : (-2.0f * cpx);
        b.y = hi ? 0.0f : (-2.0f * cpy);

        v8f c = {0.f, 0.f, 0.f, 0.f, 0.f, 0.f, 0.f, 0.f};
        // 8 args: (neg_a, A, neg_b, B, c_mod, C, reuse_a, reuse_b)
        c = __builtin_amdgcn_wmma_f32_16x16x4_f32(
            false, a, false, b, (short)0, c, false, false);

#pragma unroll
        for (int v = 0; v < 8; ++v) {
            float d2 = qq[v] + p2 + c[v];
            d2 = fmaxf(d2, 0.0f);
            const bool pred = d2 <= r2;
            const unsigned bal = __builtin_amdgcn_ballot_w32(pred);
            const unsigned halfmask = (bal >> shamt) & 0xFFFFu;
            const int off = __popc(halfmask & lanebelow);
            const int pos = cnt[v] + off;
            if (pred && pos < KCAP) {
                const unsigned o = rowoff[v] + (unsigned)pos;
                out_idx [o] = (float)j;  // exact in f32
                out_dist[o] = d2;
            }
            cnt[v] += __popc(halfmask);
        }
    }

    // Pad [count,128) with (-1, 0) and record exact counts for the scan.
#pragma unroll
    for (int v = 0; v < 8; ++v) {
        const int row = qrow_base + v + (hi ? 8 : 0);
        const int cap = cnt[v] < KCAP ? cnt[v] : KCAP;
        for (int pos = cap + col; pos < KCAP; pos += 16) {
            const unsigned o = rowoff[v] + (unsigned)pos;
            out_idx [o] = -1.0f;
            out_dist[o] = 0.0f;
        }
        if (col == 0) counts[row] = cnt[v];
    }
}

// Single-workgroup exclusive prefix sum of 8192 counts -> row_splits[8193].
__global__ __launch_bounds__(1024) void frs_scan_kernel(
    const int* __restrict__ counts, float* __restrict__ splits)
{
    __shared__ int s[1024];
    const int t = threadIdx.x;
    int local[8];
    int tot = 0;
#pragma unroll
    for (int e = 0; e < 8; ++e) {
        local[e] = counts[t * 8 + e];
        tot += local[e];
    }
    s[t] = tot;
    __syncthreads();
    // Hillis-Steele inclusive scan over 1024 thread totals.
    for (int d = 1; d < 1024; d <<= 1) {
        const int v = (t >= d) ? s[t - d] : 0;
        __syncthreads();
        s[t] += v;
        __syncthreads();
    }
    const int excl = s[t] - tot;
    if (t == 0) splits[0] = 0.0f;
    int run = excl;
#pragma unroll
    for (int e = 0; e < 8; ++e) {
        run += local[e];
        splits[t * 8 + e + 1] = (float)run;  // totals < 2^24, exact in f32
    }
}

extern "C" void kernel_launch(void* const* d_in, const int* in_sizes, int n_in,
                              void* d_out, int out_size, void* d_ws, size_t ws_size,
                              hipStream_t stream) {
    (void)in_sizes; (void)n_in; (void)out_size; (void)ws_size;
    const float* points  = (const float*)d_in[0];
    const float* queries = (const float*)d_in[1];
    const float* radius  = (const float*)d_in[2];

    float* out      = (float*)d_out;
    float* out_idx  = out;                              // [Q, 128]
    float* splits   = out + (size_t)NQRY * KCAP;        // [Q + 1]
    float* out_dist = splits + (NQRY + 1);              // [Q, 128]

    int* counts = (int*)d_ws;                           // Q ints of scratch

    // 512 tiles of 16 queries, 8 waves (tiles) per 256-thread block.
    frs_search_kernel<<<dim3(NQRY / 16 / 8), dim3(256), 0, stream>>>(
        points, queries, radius, out_idx, out_dist, counts);
    frs_scan_kernel<<<dim3(1), dim3(1024), 0, stream>>>(counts, splits);
}